// Discriminator_73598559584743
// MI455X (gfx1250) — compile-verified
//
#include <hip/hip_runtime.h>
#include <stdint.h>

// Problem constants (match reference: B=8, N=8192, D=512, fp32)
#define DIM            512
#define NROWS          8192
#define WAVES_PER_BLK  8
#define BLK_THREADS    (WAVES_PER_BLK * 32)

// ---- CDNA5 async global -> LDS copy (16 B/lane), ASYNCcnt-tracked ----
// INST_OFFSET is added to BOTH the LDS address and the global address
// (CDNA5 ISA), so one VDST/VADDR pair covers a whole row
// with immediate offsets. Non-temporal: data is streamed once (403 MB > 192 MB L2).
template <int OFS>
__device__ __forceinline__ void async_copy_b128(uint32_t lds_byte_off, const float* gptr) {
    asm volatile("global_load_async_to_lds_b128 %0, %1, off offset:%2 th:TH_LOAD_NT"
                 :
                 : "v"(lds_byte_off), "v"((unsigned long long)(uintptr_t)gptr), "n"(OFS)
                 : "memory");
}
__device__ __forceinline__ void wait_async_all() {
    asm volatile("s_wait_asynccnt 0" ::: "memory");
}

__global__ __launch_bounds__(BLK_THREADS) void cosrow_kernel(
    const float* __restrict__ s,
    const float* __restrict__ h_rl,
    const float* __restrict__ h_fk,
    float* __restrict__ out)
{
    // Per-wave staging: 3 tensors * 512 floats = 6 KB per wave, 48 KB per block.
    __shared__ float smem[WAVES_PER_BLK * 3 * DIM];

    const int lane = threadIdx.x & 31;
    const int wave = threadIdx.x >> 5;
    const int row  = blockIdx.x * WAVES_PER_BLK + wave;   // 0 .. B*N-1 (grid sized exactly)

    float* wsm = &smem[wave * 3 * DIM];
    // Flat pointer to LDS: low 32 bits == workgroup-relative LDS byte offset (ISA 10.2).
    const uint32_t lb = (uint32_t)(uintptr_t)wsm + (uint32_t)lane * 16u;

    const size_t goff = (size_t)row * DIM + (size_t)lane * 4;  // lane's first float4
    const float* ps = s    + goff;
    const float* pr = h_rl + goff;
    const float* pf = h_fk + goff;

    // 12 async b128 DMAs: wave covers 512 floats/tensor in 4 chunks of (32 lanes * 16 B),
    // chunk stride 512 B expressed via the instruction offset (applies to LDS + global).
    async_copy_b128<0>(lb, ps);             async_copy_b128<512>(lb, ps);
    async_copy_b128<1024>(lb, ps);          async_copy_b128<1536>(lb, ps);

    async_copy_b128<0>(lb + 2048u, pr);     async_copy_b128<512>(lb + 2048u, pr);
    async_copy_b128<1024>(lb + 2048u, pr);  async_copy_b128<1536>(lb + 2048u, pr);

    async_copy_b128<0>(lb + 4096u, pf);     async_copy_b128<512>(lb + 4096u, pf);
    async_copy_b128<1024>(lb + 4096u, pf);  async_copy_b128<1536>(lb + 4096u, pf);

    wait_async_all();

    // Per-lane partial sums over 16 elements per tensor (float4 LDS reads, conflict-free).
    float ss = 0.f, rr = 0.f, ff = 0.f, sr = 0.f, sf = 0.f;
#pragma unroll
    for (int c = 0; c < 4; ++c) {
        const int e = (c * 32 + lane) * 4;
        const float4 sv = *reinterpret_cast<const float4*>(&wsm[e]);
        const float4 rv = *reinterpret_cast<const float4*>(&wsm[DIM + e]);
        const float4 fv = *reinterpret_cast<const float4*>(&wsm[2 * DIM + e]);

        ss = fmaf(sv.x, sv.x, ss); ss = fmaf(sv.y, sv.y, ss);
        ss = fmaf(sv.z, sv.z, ss); ss = fmaf(sv.w, sv.w, ss);

        rr = fmaf(rv.x, rv.x, rr); rr = fmaf(rv.y, rv.y, rr);
        rr = fmaf(rv.z, rv.z, rr); rr = fmaf(rv.w, rv.w, rr);

        ff = fmaf(fv.x, fv.x, ff); ff = fmaf(fv.y, fv.y, ff);
        ff = fmaf(fv.z, fv.z, ff); ff = fmaf(fv.w, fv.w, ff);

        sr = fmaf(sv.x, rv.x, sr); sr = fmaf(sv.y, rv.y, sr);
        sr = fmaf(sv.z, rv.z, sr); sr = fmaf(sv.w, rv.w, sr);

        sf = fmaf(sv.x, fv.x, sf); sf = fmaf(sv.y, fv.y, sf);
        sf = fmaf(sv.z, fv.z, sf); sf = fmaf(sv.w, fv.w, sf);
    }

    // Wave32 butterfly reduction.
#pragma unroll
    for (int m = 16; m >= 1; m >>= 1) {
        ss += __shfl_xor(ss, m, 32);
        rr += __shfl_xor(rr, m, 32);
        ff += __shfl_xor(ff, m, 32);
        sr += __shfl_xor(sr, m, 32);
        sf += __shfl_xor(sf, m, 32);
    }

    if (lane == 0) {
        const float EPS = 1e-12f;
        const float ns = fmaxf(sqrtf(ss), EPS);
        const float nr = fmaxf(sqrtf(rr), EPS);
        const float nf = fmaxf(sqrtf(ff), EPS);
        const int b = row / NROWS;
        const int n = row - b * NROWS;
        float* orow = out + (size_t)b * (2 * NROWS);
        orow[n]         = sr / (ns * nr);
        orow[NROWS + n] = sf / (ns * nf);
    }
}

extern "C" void kernel_launch(void* const* d_in, const int* in_sizes, int n_in,
                              void* d_out, int out_size, void* d_ws, size_t ws_size,
                              hipStream_t stream) {
    (void)n_in; (void)d_ws; (void)ws_size; (void)out_size;
    const float* s    = (const float*)d_in[0];
    const float* h_rl = (const float*)d_in[1];
    const float* h_fk = (const float*)d_in[2];
    float* out = (float*)d_out;

    const int total_rows = in_sizes[0] / DIM;          // B*N = 65536
    const int grid = total_rows / WAVES_PER_BLK;       // 8192 blocks of 8 waves

    cosrow_kernel<<<grid, BLK_THREADS, 0, stream>>>(s, h_rl, h_fk, out);
}